// GlobalAttentionPoolingThenCat_17875653886195
// MI455X (gfx1250) — compile-verified
//
#include <hip/hip_runtime.h>
#include <math.h>

#ifndef __has_builtin
#define __has_builtin(x) 0
#endif

#if __has_builtin(__builtin_amdgcn_wmma_f32_16x16x4_f32)
#define USE_WMMA_F32 1
#else
#define USE_WMMA_F32 0
#endif

typedef __attribute__((ext_vector_type(2))) float v2f;
typedef __attribute__((ext_vector_type(8))) float v8f;

#define GATE_WAVES 4          // 4 waves / block -> 33.8 KB LDS (under 64 KB static cap)
#define TSTRIDE   132         // 128 + 4 pad: row readers land on banks 0,4,...,60
#define RO_NODES  256         // nodes per readout block

__device__ __forceinline__ void atomicMaxF(float* addr, float v) {
  // IEEE-754 monotone bit trick: works for mixed-sign values with -inf init.
  if (v >= 0.0f) atomicMax((int*)addr, __float_as_int(v));
  else           atomicMin((unsigned int*)addr, __float_as_uint(v));
}

// ---------------- K0a: zero pooled cols, copy global feats into tail cols ----
__global__ void init_out_kernel(const float* __restrict__ gfeat, float* __restrict__ out,
                                int G, int DG, int outW) {
  int i = blockIdx.x * blockDim.x + threadIdx.x;
  int total = G * outW;
  if (i >= total) return;
  int g = i / outW;
  int c = i - g * outW;
  int catBase = outW - DG;
  out[i] = (c < catBase) ? 0.0f : gfeat[g * DG + (c - catBase)];
}

// ---------------- K0b: max = -inf, sum = 0 ----------------------------------
__global__ void init_stats_kernel(float* __restrict__ maxA, float* __restrict__ maxB,
                                  float* __restrict__ sumA, float* __restrict__ sumB, int G) {
  int i = blockIdx.x * blockDim.x + threadIdx.x;
  if (i >= G) return;
  float ninf = __int_as_float(0xFF800000u);
  maxA[i] = ninf; maxB[i] = ninf;
  sumA[i] = 0.0f; sumB[i] = 0.0f;
}

// ---------------- K1: gate = leaky_relu(x@W + b), per-graph atomic max ------
// One wave handles a 16-node tile: coalesced float4 stage to LDS, then 32x
// accumulating V_WMMA_F32_16X16X4_F32 with W replicated across B columns.
__global__ void __launch_bounds__(GATE_WAVES * 32)
gate_max_kernel(const float* __restrict__ x, const float* __restrict__ W,
                const float* __restrict__ bias, const int* __restrict__ seg,
                float* __restrict__ gate, float* __restrict__ gmax, int N) {
  __shared__ float Wlds[128];
  __shared__ float tile[GATE_WAVES * 16 * TSTRIDE];

  const int tid  = threadIdx.x;
  const int wave = tid >> 5;
  const int lane = tid & 31;

  Wlds[tid] = W[tid];                       // blockDim.x == 128

  const int  tileIdx  = blockIdx.x * GATE_WAVES + wave;
  const int  nodeBase = tileIdx * 16;
  const bool active   = nodeBase < N;
  float* myTile = &tile[wave * 16 * TSTRIDE];

  if (active) {
#pragma unroll
    for (int row = 0; row < 16; ++row) {    // 32 lanes x float4 = one 128-f row
      int r = nodeBase + row; if (r > N - 1) r = N - 1;   // clamp tail
      float4 v = *(const float4*)(x + r * 128 + lane * 4);
      *(float4*)(myTile + row * TSTRIDE + lane * 4) = v;
    }
  }
  __syncthreads();                          // publishes Wlds + tiles
  if (!active) return;                      // wave-uniform: EXEC all-ones at WMMA

  const float bv = bias[0];

#if USE_WMMA_F32
  // A (16x4 f32): lanes 0-15 -> rows, VGPR0/1 = K+0/K+1 ; lanes 16-31 = K+2/K+3
  const int m   = lane & 15;
  const int off = (lane >> 4) << 1;         // 0 or 2
  v8f c = {0.f, 0.f, 0.f, 0.f, 0.f, 0.f, 0.f, 0.f};
#pragma unroll
  for (int k = 0; k < 128; k += 4) {
    v2f a = *(const v2f*)(myTile + m * TSTRIDE + k + off);   // 8B aligned, bank-clean
    v2f b = *(const v2f*)(Wlds + k + off);                   // W replicated over N cols
    c = __builtin_amdgcn_wmma_f32_16x16x4_f32(false, a, false, b, (short)0, c,
                                              false, false);
  }
  // C layout: lanes 0-15 hold M=0..7 (VGPR j -> M=j), lanes 16-31 hold M=8..15.
  // All columns identical (B replicated) -> lane 0 and lane 16 write 8 gates each.
  if ((lane & 15) == 0) {
    const int mbase = (lane >> 4) * 8;
#pragma unroll
    for (int j = 0; j < 8; ++j) {
      int node = nodeBase + mbase + j;
      if (node < N) {
        float g = c[j] + bv;
        g = (g > 0.f) ? g : 0.01f * g;      // leaky_relu, slope 0.01
        gate[node] = g;
        atomicMaxF(&gmax[seg[node]], g);
      }
    }
  }
#else
  // VALU fallback: lane m computes node m's dot product from LDS.
  if (lane < 16) {
    float g = 0.f;
#pragma unroll 8
    for (int k = 0; k < 128; ++k)
      g = fmaf(myTile[lane * TSTRIDE + k], Wlds[k], g);
    int node = nodeBase + lane;
    if (node < N) {
      g += bv;
      g = (g > 0.f) ? g : 0.01f * g;
      gate[node] = g;
      atomicMaxF(&gmax[seg[node]], g);
    }
  }
#endif
}

// ---------------- K2: e = exp(gate - max[seg]); sum[seg] += e ---------------
__global__ void expsum_kernel(float* __restrict__ gate, const int* __restrict__ seg,
                              const float* __restrict__ gmax, float* __restrict__ gsum,
                              int N) {
  int i = blockIdx.x * blockDim.x + threadIdx.x;
  if (i >= N) return;
  int s = seg[i];
  float e = __expf(gate[i] - gmax[s]);
  gate[i] = e;                              // overwrite gate with exp term
  atomicAdd(&gsum[s], e);
}

// ---------------- K2b: sum -> 1/sum -----------------------------------------
__global__ void invert_kernel(float* __restrict__ sumA, float* __restrict__ sumB, int G) {
  int i = blockIdx.x * blockDim.x + threadIdx.x;
  if (i < G) {
    float s = sumA[i]; sumA[i] = (s > 0.f) ? 1.f / s : 0.f;
  } else if (i < 2 * G) {
    int j = i - G;
    float s = sumB[j]; sumB[j] = (s > 0.f) ? 1.f / s : 0.f;
  }
}

// ---------------- K3: out[seg, col] += x * (e * invSum[seg]) ----------------
// 128 threads = 128 feature columns; block owns RO_NODES consecutive nodes.
// Segment ids are sorted -> register accumulation with boundary-only atomics.
__global__ void __launch_bounds__(128)
readout_kernel(const float* __restrict__ x, const float* __restrict__ eArr,
               const float* __restrict__ sumInv, const int* __restrict__ seg,
               float* __restrict__ out, int N, int outW, int colBase) {
  __shared__ int   segS[RO_NODES];
  __shared__ float aS[RO_NODES];
  const int t = threadIdx.x;
  const int s = blockIdx.x * RO_NODES;
  const int cnt = min(RO_NODES, N - s);

  for (int i = t; i < cnt; i += 128) { segS[i] = seg[s + i]; aS[i] = eArr[s + i]; }
  __syncthreads();

  float acc = 0.f;
  int   cur = segS[0];
  float inv = sumInv[cur];
  for (int i = 0; i < cnt; ++i) {
    int sg = segS[i];                       // uniform across block
    if (sg != cur) {
      atomicAdd(&out[cur * outW + colBase + t], acc);
      acc = 0.f; cur = sg; inv = sumInv[cur];
    }
    float w = aS[i] * inv;
    // streaming read, no reuse -> nontemporal hint keeps L2 for hot data
    acc = fmaf(__builtin_nontemporal_load(x + (s + i) * 128 + t), w, acc);
  }
  atomicAdd(&out[cur * outW + colBase + t], acc);
}

// ---------------------------------------------------------------------------
extern "C" void kernel_launch(void* const* d_in, const int* in_sizes, int n_in,
                              void* d_out, int out_size, void* d_ws, size_t ws_size,
                              hipStream_t stream) {
  const float* atom  = (const float*)d_in[0];
  const float* bond  = (const float*)d_in[1];
  const float* gfeat = (const float*)d_in[2];
  const int*   asg   = (const int*)d_in[3];
  const int*   bsg   = (const int*)d_in[4];
  const float* Wa    = (const float*)d_in[5];
  const float* ba    = (const float*)d_in[6];
  const float* Wb    = (const float*)d_in[7];
  const float* bb    = (const float*)d_in[8];
  float* out = (float*)d_out;

  const int D  = 128, DG = 64;
  const int NA = in_sizes[0] / D;
  const int NB = in_sizes[1] / D;
  const int G  = in_sizes[2] / DG;
  const int outW = 2 * D + DG;              // 320

  // d_ws layout (floats): gateA[NA] gateB[NB] maxA[G] maxB[G] sumA[G] sumB[G]
  float* ws    = (float*)d_ws;
  float* gateA = ws;
  float* gateB = gateA + NA;
  float* maxA  = gateB + NB;
  float* maxB  = maxA + G;
  float* sumA  = maxB + G;
  float* sumB  = sumA + G;

  {
    int total = G * outW;
    init_out_kernel<<<(total + 255) / 256, 256, 0, stream>>>(gfeat, out, G, DG, outW);
    init_stats_kernel<<<(G + 255) / 256, 256, 0, stream>>>(maxA, maxB, sumA, sumB, G);
  }

  {
    int tilesA = (NA + 15) / 16;
    int tilesB = (NB + 15) / 16;
    gate_max_kernel<<<(tilesA + GATE_WAVES - 1) / GATE_WAVES, GATE_WAVES * 32, 0, stream>>>(
        atom, Wa, ba, asg, gateA, maxA, NA);
    gate_max_kernel<<<(tilesB + GATE_WAVES - 1) / GATE_WAVES, GATE_WAVES * 32, 0, stream>>>(
        bond, Wb, bb, bsg, gateB, maxB, NB);
  }

  expsum_kernel<<<(NA + 255) / 256, 256, 0, stream>>>(gateA, asg, maxA, sumA, NA);
  expsum_kernel<<<(NB + 255) / 256, 256, 0, stream>>>(gateB, bsg, maxB, sumB, NB);

  invert_kernel<<<(2 * G + 255) / 256, 256, 0, stream>>>(sumA, sumB, G);

  readout_kernel<<<(NA + RO_NODES - 1) / RO_NODES, 128, 0, stream>>>(
      atom, gateA, sumA, asg, out, NA, outW, 0);
  readout_kernel<<<(NB + RO_NODES - 1) / RO_NODES, 128, 0, stream>>>(
      bond, gateB, sumB, bsg, out, NB, outW, D);
}